// Net_63720134803892
// MI455X (gfx1250) — compile-verified
//
#include <hip/hip_runtime.h>
#include <hip/hip_bf16.h>

typedef __attribute__((ext_vector_type(2))) float v2f;
typedef __attribute__((ext_vector_type(8))) float v8f;

// Problem constants (match reference)
constexpr int B_GR    = 64;
constexpr int N_GENES = 16000;
constexpr int N_NODES = 10000;
constexpr int N_ANNOT = 8000;
constexpr int N_TOTAL = B_GR * N_NODES;
constexpr int K_KEEP  = 5000;      // ceil(0.5 * N_NODES)

// GEMM tiling
constexpr int KSTEP = 32;          // K panel per LDS stage (8 WMMA k=4 steps)
constexpr int LSTR  = 36;          // LDS row stride in floats (16B-aligned rows, bank-conflict-free)

// ---------------------------------------------------------------------------
// Phase 1: emb[64,8000] = T[64,16000] @ (W*Mask)^T + bias   (HBM-bound, WMMA)
// Block: 128 threads = 4 waves; wave w computes M-tile w (rows 16w..16w+15),
// all waves share one 16-column B-panel in LDS -> weights read exactly once.
// ---------------------------------------------------------------------------
__global__ __launch_bounds__(128) void fc1_wmma_kernel(
    const float* __restrict__ T,     // [64, 16000]
    const float* __restrict__ W,     // [8000, 16000]
    const float* __restrict__ Mk,    // [8000, 16000] mask
    const float* __restrict__ bias,  // [8000]
    float* __restrict__ emb)         // [64, 8000]
{
    __shared__ float ldsA[64 * LSTR];   // A panel: 64 m x 32 k
    __shared__ float ldsB[16 * LSTR];   // B panel: 16 n x 32 k (masked weights)

    const int n0    = blockIdx.x * 16;       // 500 blocks cover N=8000
    const int t     = threadIdx.x;
    const int lane  = t & 31;
    const int wave  = t >> 5;                // == M-tile index 0..3
    const int laneM = lane & 15;
    const int half  = lane >> 4;             // 0: K=0,1   1: K=2,3

    v8f acc = {};

    const int nB = t >> 3;                   // B-panel row this thread loads
    const int jB = (t & 7) * 4;              // float4 offset within the row

    for (int k0 = 0; k0 < N_GENES; k0 += KSTEP) {
        __syncthreads();   // previous iteration's LDS reads done

        // --- B panel: masked weights, 16 rows x 32 k, one float4/thread/array
        {
            const size_t base = (size_t)(n0 + nB) * N_GENES + k0 + jB;
            float4 w4 = *(const float4*)(W  + base);
            float4 m4 = *(const float4*)(Mk + base);
            float4 p;
            p.x = w4.x * m4.x; p.y = w4.y * m4.y;
            p.z = w4.z * m4.z; p.w = w4.w * m4.w;
            *(float4*)(&ldsB[nB * LSTR + jB]) = p;
            // prefetch next K panel of the streamed weights
            if (k0 + KSTEP < N_GENES) {
                __builtin_prefetch(W  + base + KSTEP, 0, 0);
                __builtin_prefetch(Mk + base + KSTEP, 0, 0);
            }
        }
        // --- A panel: 64 rows x 32 k, 4 float4 per thread
        #pragma unroll
        for (int r = 0; r < 4; ++r) {
            int idx = t + 128 * r;
            int m = idx >> 3, j = (idx & 7) * 4;
            *(float4*)(&ldsA[m * LSTR + j]) =
                *(const float4*)(T + (size_t)m * N_GENES + k0 + j);
        }
        __syncthreads();

        // --- 8 x V_WMMA_F32_16X16X4_F32
        #pragma unroll
        for (int ks = 0; ks < KSTEP; ks += 4) {
            const int kk = ks + 2 * half;
            v2f a, b;
            a.x = ldsA[(wave * 16 + laneM) * LSTR + kk];      // A[m][k]
            a.y = ldsA[(wave * 16 + laneM) * LSTR + kk + 1];
            b.x = ldsB[laneM * LSTR + kk];                    // B[k][n] = W[n][k]*M[n][k]
            b.y = ldsB[laneM * LSTR + kk + 1];
            acc = __builtin_amdgcn_wmma_f32_16x16x4_f32(
                false, a, false, b, (short)0, acc, false, false);
        }
    }

    // C/D layout: VGPR r -> (M = r + 8*half, N = laneM)
    const float bn = bias[n0 + laneM];
    #pragma unroll
    for (int r = 0; r < 8; ++r) {
        int m = wave * 16 + r + 8 * half;
        emb[(size_t)m * N_ANNOT + n0 + laneM] = acc[r] + bn;
    }
}

// ---------------------------------------------------------------------------
// Phase 2: xg = x with first N_ANNOT nodes of each graph replaced by emb
// ---------------------------------------------------------------------------
__global__ void build_xg_kernel(const float* __restrict__ x,
                                const float* __restrict__ emb,
                                float* __restrict__ xg)
{
    int i = blockIdx.x * blockDim.x + threadIdx.x;
    if (i >= N_TOTAL) return;
    int b = i / N_NODES, n = i - b * N_NODES;
    xg[i] = (n < N_ANNOT) ? emb[b * N_ANNOT + n] : x[i];
}

__global__ void zero_kernel(float* __restrict__ p, int n)
{
    int i = blockIdx.x * blockDim.x + threadIdx.x;
    if (i < n) p[i] = 0.0f;
}

// ---------------------------------------------------------------------------
// Phase 3: edge scatter (msg_sum, degree) with f32 global atomics
// ---------------------------------------------------------------------------
__global__ void edge_kernel(const long long* __restrict__ ei,  // [2, nE] int64
                            const float* __restrict__ xg,
                            float* __restrict__ msg,
                            float* __restrict__ deg,
                            long long nE)
{
    long long e = (long long)blockIdx.x * blockDim.x + threadIdx.x;
    if (e >= nE) return;
    long long s = ei[e];
    long long d = ei[nE + e];
    atomicAdd(&msg[d], xg[s]);
    atomicAdd(&deg[d], 1.0f);
}

// ---------------------------------------------------------------------------
// Phase 4: h = w_root*x + w_nbr*(msg/max(deg,1)) + b
// ---------------------------------------------------------------------------
__global__ void prop_kernel(const float* __restrict__ xg,
                            const float* __restrict__ msg,
                            const float* __restrict__ deg,
                            const float* __restrict__ w_root,
                            const float* __restrict__ w_nbr,
                            const float* __restrict__ pb,
                            float* __restrict__ h)
{
    int i = blockIdx.x * blockDim.x + threadIdx.x;
    if (i >= N_TOTAL) return;
    float mean = msg[i] / fmaxf(deg[i], 1.0f);
    h[i] = w_root[0] * xg[i] + w_nbr[0] * mean + pb[0];
}

// ---------------------------------------------------------------------------
// Phase 5a: per-graph radix select on IEEE bits of |h| -> k-th largest value
// ---------------------------------------------------------------------------
__global__ void topk_thresh_kernel(const float* __restrict__ h,
                                   unsigned* __restrict__ tbits)
{
    const int g = blockIdx.x;
    const float* hg = h + (size_t)g * N_NODES;
    __shared__ unsigned hist[256];
    __shared__ unsigned s_prefix, s_k;
    if (threadIdx.x == 0) { s_prefix = 0u; s_k = K_KEEP; }
    __syncthreads();

    for (int pass = 0; pass < 4; ++pass) {
        const int shift = 24 - pass * 8;
        const unsigned mask_hi = (pass == 0) ? 0u : (0xFFFFFFFFu << (shift + 8));
        for (int i = threadIdx.x; i < 256; i += blockDim.x) hist[i] = 0u;
        __syncthreads();
        const unsigned prefix = s_prefix;
        for (int i = threadIdx.x; i < N_NODES; i += blockDim.x) {
            unsigned bits = __float_as_uint(fabsf(hg[i]));   // monotone for >=0
            if ((bits & mask_hi) == (prefix & mask_hi))
                atomicAdd(&hist[(bits >> shift) & 0xFF], 1u);
        }
        __syncthreads();
        if (threadIdx.x == 0) {
            unsigned k = s_k;
            for (int b = 255; b >= 0; --b) {
                unsigned c = hist[b];
                if (k <= c) { s_prefix = prefix | ((unsigned)b << shift); s_k = k; break; }
                k -= c;
            }
        }
        __syncthreads();
    }
    if (threadIdx.x == 0) tbits[g] = s_prefix;
}

// ---------------------------------------------------------------------------
// Phase 5b: out[b,c] = sum_i (|h|>=t ? h : 0) * fc2_w[c,i] + fc2_b[c]
// ---------------------------------------------------------------------------
__global__ void out_kernel(const float* __restrict__ h,
                           const unsigned* __restrict__ tbits,
                           const float* __restrict__ fc2_w,
                           const float* __restrict__ fc2_b,
                           float* __restrict__ out)
{
    const int g = blockIdx.x;
    const unsigned t = tbits[g];
    const float* hg = h + (size_t)g * N_NODES;
    float s0 = 0.f, s1 = 0.f;
    for (int i = threadIdx.x; i < N_NODES; i += blockDim.x) {
        float v = hg[i];
        if (__float_as_uint(fabsf(v)) >= t) {
            s0 += v * fc2_w[i];
            s1 += v * fc2_w[N_NODES + i];
        }
    }
    __shared__ float r0[256], r1[256];
    r0[threadIdx.x] = s0; r1[threadIdx.x] = s1;
    __syncthreads();
    for (int off = 128; off > 0; off >>= 1) {
        if ((int)threadIdx.x < off) {
            r0[threadIdx.x] += r0[threadIdx.x + off];
            r1[threadIdx.x] += r1[threadIdx.x + off];
        }
        __syncthreads();
    }
    if (threadIdx.x == 0) {
        out[g * 2 + 0] = r0[0] + fc2_b[0];
        out[g * 2 + 1] = r1[0] + fc2_b[1];
    }
}

// ---------------------------------------------------------------------------
extern "C" void kernel_launch(void* const* d_in, const int* in_sizes, int n_in,
                              void* d_out, int out_size, void* d_ws, size_t ws_size,
                              hipStream_t stream)
{
    const float*     T     = (const float*)d_in[0];      // transcriptomic [64,16000]
    const float*     x     = (const float*)d_in[1];      // [N_TOTAL,1]
    const long long* ei    = (const long long*)d_in[2];  // edge_index int64 [2,nE]
    //               batch  = d_in[3] (unused: graphs are contiguous blocks)
    const float*     adj   = (const float*)d_in[4];      // [8000,16000]
    const float*     fc1w  = (const float*)d_in[5];      // [8000,16000]
    const float*     fc1b  = (const float*)d_in[6];      // [8000]
    const float*     wroot = (const float*)d_in[7];
    const float*     wnbr  = (const float*)d_in[8];
    const float*     pb    = (const float*)d_in[9];
    const float*     fc2w  = (const float*)d_in[10];     // [2,10000]
    const float*     fc2b  = (const float*)d_in[11];     // [2]
    float*           out   = (float*)d_out;

    char* ws = (char*)d_ws;
    float*    emb   = (float*)ws;                  ws += sizeof(float) * (size_t)B_GR * N_ANNOT;
    float*    xg    = (float*)ws;                  ws += sizeof(float) * (size_t)N_TOTAL;
    float*    msg   = (float*)ws;                  ws += sizeof(float) * (size_t)N_TOTAL;
    float*    deg   = (float*)ws;                  ws += sizeof(float) * (size_t)N_TOTAL;
    float*    h     = (float*)ws;                  ws += sizeof(float) * (size_t)N_TOTAL;
    unsigned* tbits = (unsigned*)ws;

    const long long nE = (long long)in_sizes[2] / 2;

    // 1) masked fc1 GEMM (WMMA, HBM-bound)
    fc1_wmma_kernel<<<N_ANNOT / 16, 128, 0, stream>>>(T, fc1w, adj, fc1b, emb);

    // 2) node features
    build_xg_kernel<<<(N_TOTAL + 255) / 256, 256, 0, stream>>>(x, emb, xg);

    // 3) graph conv: zero accumulators then edge scatter
    zero_kernel<<<(N_TOTAL + 255) / 256, 256, 0, stream>>>(msg, N_TOTAL);
    zero_kernel<<<(N_TOTAL + 255) / 256, 256, 0, stream>>>(deg, N_TOTAL);
    edge_kernel<<<(int)((nE + 255) / 256), 256, 0, stream>>>(ei, xg, msg, deg, nE);

    // 4) propagate
    prop_kernel<<<(N_TOTAL + 255) / 256, 256, 0, stream>>>(xg, msg, deg, wroot, wnbr, pb, h);

    // 5) per-graph top-k threshold + masked fc2 reduction
    topk_thresh_kernel<<<B_GR, 256, 0, stream>>>(h, tbits);
    out_kernel<<<B_GR, 256, 0, stream>>>(h, tbits, fc2w, fc2b, out);
}